// HashGrid_79164837200474
// MI455X (gfx1250) — compile-verified
//
#include <hip/hip_runtime.h>
#include <hip/hip_bf16.h>

// HashGrid encode for MI455X (gfx1250, wave32).
// 16 lanes per pixel (lane = feature index f in 0..15). Each hash-table row is
// 64B-aligned 64B, so every corner gather is one coalesced cacheline request,
// and every per-level output store is one coalesced 64B burst.
//
// CDNA5 path: corner gathers are issued with GLOBAL_LOAD_ASYNC_TO_LDS_B32
// (tracked by ASYNCcnt, in-order completion) in a 2-deep software pipeline:
// while level l is interpolated out of LDS, level l+1's 8 corner rows are
// already streaming in. s_wait_asynccnt 0x8 with <=16 outstanding ops
// guarantees the previous level's batch has landed. Each wave owns a private
// 2KB double buffer -> no barriers.

#define L_LEVELS 16
#define T_SIZE   65536
#define F_FEAT   16
#define OUT_F    295   // 16*16 feat + 3 xt + 36 pos-enc

__global__ __launch_bounds__(256) void hashgrid_enc_kernel(
    const float* __restrict__ x,       // (P, 3)
    const float* __restrict__ t,       // (N, 1)
    const float* __restrict__ tables,  // (3, T, F)
    const int*   __restrict__ mask,    // (3,)
    float* __restrict__ out,           // (P, 295)
    int numPixels, int hw)
{
    // per-wave double buffer: [wave][parity][lane][corner]
    __shared__ __align__(16) float asbuf[8][2][32][8];

    const unsigned tid  = blockIdx.x * blockDim.x + threadIdx.x;
    const unsigned pix  = tid >> 4;          // 16 lanes per pixel
    const unsigned f    = tid & 15u;         // feature lane
    const unsigned lane = threadIdx.x & 31u; // lane in wave32
    const unsigned wv   = threadIdx.x >> 5;  // wave in block
    if (pix >= (unsigned)numPixels) return;

    // ----- resolve keep/drop axes from mask (stable argsort of mask==0) -----
    const int m0 = mask[0], m1 = mask[1], m2 = mask[2];
    int a = ((m0 == 0) ? 4 : 0) + 0;
    int b = ((m1 == 0) ? 4 : 0) + 1;
    int c = ((m2 == 0) ? 4 : 0) + 2;
    int tmp;
    if (a > b) { tmp = a; a = b; b = tmp; }
    if (b > c) { tmp = b; b = c; c = tmp; }
    if (a > b) { tmp = a; a = b; b = tmp; }
    const int keep0 = a & 3, keep1 = b & 3, drop = c & 3;

    // ----- build xt = (x[keep0], x[keep1], t) -----
    const float xv0 = x[pix * 3 + 0];
    const float xv1 = x[pix * 3 + 1];
    const float xv2 = x[pix * 3 + 2];
    const unsigned n = pix / (unsigned)hw;
    const float tv = t[n];
    const float xt0 = (keep0 == 0) ? xv0 : ((keep0 == 1) ? xv1 : xv2);
    const float xt1 = (keep1 == 0) ? xv0 : ((keep1 == 1) ? xv1 : xv2);
    const float xt2 = tv;

    const float* __restrict__ table = tables + (size_t)drop * (T_SIZE * F_FEAT);
    float* __restrict__ orow = out + (size_t)pix * OUT_F;

    // NL[l] = floor(16 * 2^(l/3))
    const float NL[L_LEVELS] = {16.f, 20.f, 25.f, 32.f, 40.f, 50.f, 64.f, 80.f,
                                101.f, 128.f, 161.f, 203.f, 256.f, 322.f, 406.f, 512.f};

    // LDS byte offsets of this lane's two corner buffers (low 32 bits of the
    // flat address of an LDS location == its LDS byte offset).
    const unsigned ldsA = (unsigned)(uintptr_t)(&asbuf[wv][0][lane][0]);
    const unsigned ldsB = (unsigned)(uintptr_t)(&asbuf[wv][1][lane][0]);

    // ---- async-issue the 8 corner-row gathers of level `l` into LDS ----
    auto issue_level = [&](int l, unsigned ldsBase) {
        const float s0 = xt0 * NL[l];
        const float s1 = xt1 * NL[l];
        const float s2 = xt2 * NL[l];
        const unsigned lo0 = (unsigned)(int)floorf(s0);
        const unsigned lo1 = (unsigned)(int)floorf(s1);
        const unsigned lo2 = (unsigned)(int)floorf(s2);
        const unsigned up0 = (unsigned)(int)ceilf(s0);
        const unsigned up1 = (unsigned)(int)ceilf(s1);
        const unsigned up2 = (unsigned)(int)ceilf(s2);
        const unsigned h1l = lo1 * 2654435761u, h1u = up1 * 2654435761u;
        const unsigned h2l = lo2 * 805459861u,  h2u = up2 * 805459861u;
        unsigned idx[8];
        idx[0] = (lo0 ^ h1l ^ h2l) & 0xFFFFu;
        idx[1] = (lo0 ^ h1l ^ h2u) & 0xFFFFu;
        idx[2] = (lo0 ^ h1u ^ h2l) & 0xFFFFu;
        idx[3] = (lo0 ^ h1u ^ h2u) & 0xFFFFu;
        idx[4] = (up0 ^ h1l ^ h2l) & 0xFFFFu;
        idx[5] = (up0 ^ h1l ^ h2u) & 0xFFFFu;
        idx[6] = (up0 ^ h1u ^ h2l) & 0xFFFFu;
        idx[7] = (up0 ^ h1u ^ h2u) & 0xFFFFu;
        #pragma unroll
        for (int cc = 0; cc < 8; ++cc) {
            const unsigned voff = idx[cc] * (F_FEAT * 4u) + f * 4u; // byte off in table
            const unsigned loff = ldsBase + (unsigned)cc * 4u;      // byte off in LDS
            // GVS form: LDS-dest VGPR, 32-bit byte-offset VGPR, SGPR-pair base
            asm volatile("global_load_async_to_lds_b32 %0, %1, %2"
                         :: "v"(loff), "v"(voff), "s"(table) : "memory");
        }
    };

    issue_level(0, ldsA);

    #pragma unroll
    for (int l = 0; l < L_LEVELS; ++l) {
        if (l + 1 < L_LEVELS) {
            issue_level(l + 1, (l & 1) ? ldsA : ldsB);
            // <=16 outstanding; in-order completion => level l's 8 are done
            asm volatile("s_wait_asynccnt 0x8" ::: "memory");
        } else {
            asm volatile("s_wait_asynccnt 0x0" ::: "memory");
        }

        // weights for level l (cheap recompute)
        const float s0 = xt0 * NL[l];
        const float s1 = xt1 * NL[l];
        const float s2 = xt2 * NL[l];
        const float w0 = s0 - floorf(s0);
        const float w1 = s1 - floorf(s1);
        const float w2 = s2 - floorf(s2);

        // read back this lane's 8 corner values: 2x ds_load_b128
        const float4* rb = reinterpret_cast<const float4*>(&asbuf[wv][l & 1][lane][0]);
        const float4 q0 = rb[0];
        const float4 q1 = rb[1];

        const float wx0 = 1.f - w0, wy0 = 1.f - w1, wz0 = 1.f - w2;
        const float wy0z0 = wy0 * wz0, wy0z1 = wy0 * w2;
        const float wy1z0 = w1 * wz0,  wy1z1 = w1 * w2;
        float acc;
        acc = q0.x * (wx0 * wy0z0);
        acc = fmaf(q0.y, wx0 * wy0z1, acc);
        acc = fmaf(q0.z, wx0 * wy1z0, acc);
        acc = fmaf(q0.w, wx0 * wy1z1, acc);
        acc = fmaf(q1.x, w0 * wy0z0, acc);
        acc = fmaf(q1.y, w0 * wy0z1, acc);
        acc = fmaf(q1.z, w0 * wy1z0, acc);
        acc = fmaf(q1.w, w0 * wy1z1, acc);

        // streaming (non-temporal) store: 16 lanes -> one 64B burst
        __builtin_nontemporal_store(acc, orow + l * F_FEAT + f);
    }

    // ----- positional encoding: 39 values, covered by lanes in 3 rounds -----
    // layout: [xt0, xt1, xt2, (sin j0 d0..d2, cos j0 d0..d2), j=0..5]
    #pragma unroll
    for (int r = 0; r < 3; ++r) {
        const int v = (int)f + r * 16;
        if (v < 39) {
            float val;
            if (v < 3) {
                val = (v == 0) ? xt0 : ((v == 1) ? xt1 : xt2);
            } else {
                const int u  = v - 3;
                const int j  = u / 6;        // frequency octave
                const int rr = u % 6;        // 0..2 -> sin, 3..5 -> cos
                const int d  = rr % 3;       // dimension
                const float xd = (d == 0) ? xt0 : ((d == 1) ? xt1 : xt2);
                const float ang = xd * (3.14159265358979323846f * (float)(1 << j));
                val = (rr < 3) ? __sinf(ang) : __cosf(ang);
            }
            __builtin_nontemporal_store(val, orow + 256 + v);
        }
    }
}

extern "C" void kernel_launch(void* const* d_in, const int* in_sizes, int n_in,
                              void* d_out, int out_size, void* d_ws, size_t ws_size,
                              hipStream_t stream) {
    const float* x      = (const float*)d_in[0];
    const float* t      = (const float*)d_in[1];
    const float* tables = (const float*)d_in[2];
    const int*   mask   = (const int*)d_in[3];
    float* out = (float*)d_out;

    const int pixels = in_sizes[0] / 3;     // N*H*W
    const int nimg   = in_sizes[1];         // N (t has one scalar per image)
    const int hw     = pixels / nimg;       // H*W

    const int totalThreads = pixels * 16;   // 16 lanes per pixel
    dim3 block(256);
    dim3 grid((totalThreads + 255) / 256);
    hipLaunchKernelGGL(hashgrid_enc_kernel, grid, block, 0, stream,
                       x, t, tables, mask, out, pixels, hw);
}